// VecDivIteration_29875792511513
// MI455X (gfx1250) — compile-verified
//
#include <hip/hip_runtime.h>
#include <hip/hip_bf16.h>

// VecDivIteration: one restoring-division step on bit-vector operands.
// Rows are 32-bit integers encoded as 0/1 floats (LSB first).
//   trial  = (r - d) mod 2^32 ; borrow = (r < d)
//   Q      = 1 - borrow       ; R_next = borrow ? r : trial
// Memory-bound: 388 B/row streamed -> ~16.6 us floor at 23.3 TB/s.
// CDNA5 path used: GLOBAL_LOAD_ASYNC_TO_LDS_B128 staging + s_wait_asynccnt.

#ifndef __has_builtin
#define __has_builtin(x) 0
#endif

#if __has_builtin(__builtin_amdgcn_global_load_async_to_lds_b128)
#define ATHENA_ASYNC_BUILTIN 1
#else
#define ATHENA_ASYNC_BUILTIN 0
#endif

#if __has_builtin(__builtin_amdgcn_s_wait_asynccnt)
#define ATHENA_WAITASYNC_BUILTIN 1
#else
#define ATHENA_WAITASYNC_BUILTIN 0
#endif

#define BITS 32
#define TILE 128   // rows per block == threads per block (4 waves of 32)
#define ROWF 36    // padded LDS row stride in floats (144 B -> <=2-way bank conflicts)

// 16-byte int vector type expected by the async-LDS builtins.
typedef int athena_v4i __attribute__((ext_vector_type(4)));
typedef __attribute__((address_space(1))) athena_v4i* athena_gptr;
typedef __attribute__((address_space(3))) athena_v4i* athena_lptr;

// Copy 16 bytes global -> LDS asynchronously (tracked by ASYNCcnt).
__device__ __forceinline__ void async_copy_b128(const void* gsrc, void* ldst) {
#if ATHENA_ASYNC_BUILTIN
  __builtin_amdgcn_global_load_async_to_lds_b128(
      (athena_gptr)(void*)gsrc,
      (athena_lptr)ldst,
      /*offset=*/0, /*cpol=*/0);
#else
  unsigned loff = (unsigned)(unsigned long long)
      (__attribute__((address_space(3))) char*)ldst;
  asm volatile("global_load_async_to_lds_b128 %0, %1, off"
               :
               : "v"(loff), "v"(gsrc)
               : "memory");
#endif
}

__device__ __forceinline__ void wait_async0() {
#if ATHENA_WAITASYNC_BUILTIN
  __builtin_amdgcn_s_wait_asynccnt(0);
#else
  asm volatile("s_wait_asynccnt 0x0" ::: "memory");
#endif
}

// Inputs are exactly 0.0f (0x00000000) or 1.0f (0x3F800000): bit 29 selects.
__device__ __forceinline__ unsigned bit01(float f) {
  return (__float_as_uint(f) >> 29) & 1u;
}

__global__ __launch_bounds__(TILE) void VecDivIteration_kernel(
    const float* __restrict__ R, const float* __restrict__ D,
    float* __restrict__ out, int n) {
  __shared__ __align__(16) float sR[TILE * ROWF];
  __shared__ __align__(16) float sD[TILE * ROWF];

  const int t = threadIdx.x;
  const long long row0 = (long long)blockIdx.x * TILE;
  const long long row = row0 + t;

  float* __restrict__ qout = out;                 // Q_bit: n floats
  float* __restrict__ rout = out + (long long)n;  // R_next: n*32 floats

  unsigned ur = 0u, ud = 0u;

  if (row0 + TILE <= (long long)n) {
    // ---- Stage full 2 x 16 KB tile into LDS with async b128 copies ----
    const float* gR = R + row0 * BITS;
    const float* gD = D + row0 * BITS;
#pragma unroll
    for (int k = 0; k < 8; ++k) {
      const int c = t + k * TILE;  // 16-byte chunk id within the tile (coalesced)
      const unsigned loff =
          (unsigned)(c >> 3) * (ROWF * 4u) + (unsigned)(c & 7) * 16u;
      async_copy_b128(gR + c * 4, (char*)sR + loff);
      async_copy_b128(gD + c * 4, (char*)sD + loff);
    }
    wait_async0();
    __syncthreads();

    // ---- Pack this thread's row from LDS (ds_load_b128, padded rows) ----
    const float4* r4 = (const float4*)(sR + t * ROWF);
    const float4* d4 = (const float4*)(sD + t * ROWF);
#pragma unroll
    for (int j = 0; j < 8; ++j) {
      const float4 a = r4[j];
      const float4 b = d4[j];
      const int s = 4 * j;
      ur |= (bit01(a.x) << s) | (bit01(a.y) << (s + 1)) |
            (bit01(a.z) << (s + 2)) | (bit01(a.w) << (s + 3));
      ud |= (bit01(b.x) << s) | (bit01(b.y) << (s + 1)) |
            (bit01(b.z) << (s + 2)) | (bit01(b.w) << (s + 3));
    }
  } else {
    // Tail block (not hit at N=2^20): direct loads, no barriers used here.
    if (row >= (long long)n) return;
    const float* rr = R + row * BITS;
    const float* dd = D + row * BITS;
#pragma unroll
    for (int i = 0; i < BITS; ++i) {
      ur |= bit01(rr[i]) << i;
      ud |= bit01(dd[i]) << i;
    }
  }

  // ---- One restoring-division step in integer domain ----
  const unsigned trial = ur - ud;       // (r - d) mod 2^32
  const bool borrow = (ur < ud);        // final ripple borrow
  const unsigned sel = borrow ? ur : trial;

  qout[row] = borrow ? 0.0f : 1.0f;     // Q = NOT(borrow)

  float* o = rout + row * BITS;
#pragma unroll
  for (int j = 0; j < 8; ++j) {
    float4 v;
    v.x = (float)((sel >> (4 * j + 0)) & 1u);
    v.y = (float)((sel >> (4 * j + 1)) & 1u);
    v.z = (float)((sel >> (4 * j + 2)) & 1u);
    v.w = (float)((sel >> (4 * j + 3)) & 1u);
    ((float4*)o)[j] = v;                // global_store_b128
  }
}

extern "C" void kernel_launch(void* const* d_in, const int* in_sizes, int n_in,
                              void* d_out, int out_size, void* d_ws,
                              size_t ws_size, hipStream_t stream) {
  (void)n_in; (void)out_size; (void)d_ws; (void)ws_size;
  const float* R = (const float*)d_in[0];
  const float* D = (const float*)d_in[1];
  float* out = (float*)d_out;
  const int n = in_sizes[0] / BITS;  // number of rows (1048576)
  const int grid = (n + TILE - 1) / TILE;
  VecDivIteration_kernel<<<grid, TILE, 0, stream>>>(R, D, out, n);
}